// DbrxRouter_40492951667584
// MI455X (gfx1250) — compile-verified
//
#include <hip/hip_runtime.h>
#include <hip/hip_bf16.h>

typedef float v2f __attribute__((ext_vector_type(2)));
typedef float v8f __attribute__((ext_vector_type(8)));

#define HDIM 6144
#define NEXP 16
#define TOPK 4
#define WAVES_PER_BLOCK 8

// One wave computes a 16-token x 16-expert logits tile with V_WMMA_F32_16X16X4_F32,
// then lanes 0..15 each finish one token (softmax-free top-4 renorm).
__global__ __launch_bounds__(32 * WAVES_PER_BLOCK)
void dbrx_router_wmma(const float* __restrict__ x, const float* __restrict__ w,
                      float* __restrict__ out_w, int* __restrict__ out_e,
                      int ntiles)
{
    __shared__ float tileC[WAVES_PER_BLOCK][16][NEXP + 1];  // +1 pad: bank-conflict-free

    const int lane = threadIdx.x & 31;
    const int wave = threadIdx.x >> 5;
    const int tile = blockIdx.x * WAVES_PER_BLOCK + wave;
    if (tile >= ntiles) return;  // uniform per wave

    // 32-bit WMMA A(16x4) fragment layout: lanes 0-15 -> K={0,1}, lanes 16-31 -> K={2,3}
    const int r  = lane & 15;           // token row (A) / expert col (B)
    const int kh = (lane >> 4) << 1;    // 0 or 2

    const float* xp = x + (size_t)(tile * 16 + r) * HDIM + kh;
    const float* wp = w + (size_t)r * HDIM + kh;

    v8f acc0 = {0.f, 0.f, 0.f, 0.f, 0.f, 0.f, 0.f, 0.f};
    v8f acc1 = {0.f, 0.f, 0.f, 0.f, 0.f, 0.f, 0.f, 0.f};

    #pragma unroll 4
    for (int k = 0; k < HDIM; k += 8) {
        // x is streamed exactly once -> non-temporal; W is L2-hot (reused 512x) -> regular.
        v2f a0 = __builtin_nontemporal_load((const v2f*)(xp + k));
        v2f a1 = __builtin_nontemporal_load((const v2f*)(xp + k + 4));
        v2f b0 = *(const v2f*)(wp + k);
        v2f b1 = *(const v2f*)(wp + k + 4);
        // 8 args: (neg_a, A, neg_b, B, c_mod, C, reuse_a, reuse_b)
        acc0 = __builtin_amdgcn_wmma_f32_16x16x4_f32(false, a0, false, b0,
                                                     (short)0, acc0, false, false);
        acc1 = __builtin_amdgcn_wmma_f32_16x16x4_f32(false, a1, false, b1,
                                                     (short)0, acc1, false, false);
    }

    // C/D layout: VGPR v -> row (v + 8*(lane>=16)), col = lane&15. Spill tile to LDS.
    const int cn = lane & 15;
    const int mb = (lane >> 4) << 3;
    #pragma unroll
    for (int v = 0; v < 8; ++v)
        tileC[wave][mb + v][cn] = acc0[v] + acc1[v];

    __syncthreads();

    // Lanes 0..15: one token each. top-4 of logits == top-4 of softmax (monotone),
    // and p=1 renorm cancels the softmax denominator:
    //   w_j = exp(l_j - m) / sum_{top4} exp(l_i - m)
    if (lane < 16) {
        float lg[NEXP];
        #pragma unroll
        for (int e = 0; e < NEXP; ++e) lg[e] = tileC[wave][lane][e];

        float m = lg[0];
        #pragma unroll
        for (int e = 1; e < NEXP; ++e) m = fmaxf(m, lg[e]);

        unsigned used = 0u;
        float pw[TOPK];
        int   pe[TOPK];
        float s = 0.f;
        #pragma unroll
        for (int j = 0; j < TOPK; ++j) {
            float bv = -3.402823466e38f;
            int   bi = 0;
            #pragma unroll
            for (int e = 0; e < NEXP; ++e) {
                bool ok = (((used >> e) & 1u) == 0u) && (lg[e] > bv);
                bv = ok ? lg[e] : bv;
                bi = ok ? e : bi;   // strict '>' keeps lowest index on ties (jax top_k)
            }
            used |= (1u << bi);
            float p = __expf(bv - m);
            pw[j] = p;
            pe[j] = bi;
            s += p;
        }
        const float inv = 1.0f / s;

        const size_t t = (size_t)tile * 16 + lane;
        #pragma unroll
        for (int j = 0; j < TOPK; ++j) {
            out_w[t * TOPK + j] = pw[j] * inv;
            out_e[t * TOPK + j] = pe[j];
        }
    }
}

extern "C" void kernel_launch(void* const* d_in, const int* in_sizes, int n_in,
                              void* d_out, int out_size, void* d_ws, size_t ws_size,
                              hipStream_t stream) {
    const float* x = (const float*)d_in[0];   // [4,2048,6144] f32
    const float* w = (const float*)d_in[1];   // [16,6144]    f32

    const int T = in_sizes[0] / HDIM;         // 8192 tokens
    float* out_w = (float*)d_out;                             // [T,4] f32
    int*   out_e = (int*)((float*)d_out + (size_t)T * TOPK);  // [T,4] i32, concatenated

    const int ntiles = T / 16;                                // 512
    const int blocks = (ntiles + WAVES_PER_BLOCK - 1) / WAVES_PER_BLOCK;  // 64

    dbrx_router_wmma<<<blocks, 32 * WAVES_PER_BLOCK, 0, stream>>>(x, w, out_w, out_e, ntiles);
}